// TopKGate_13159779794953
// MI455X (gfx1250) — compile-verified
//
#include <hip/hip_runtime.h>
#include <hip/hip_bf16.h>

// ---------------- problem constants (match the reference) ----------------
constexpr int S   = 8192;   // tokens
constexpr int MD  = 4096;   // model dim
constexpr int NE  = 64;     // experts
constexpr int CAP = 128;    // capacity
constexpr long long SEC = (long long)S * NE * CAP;  // 67,108,864

typedef __attribute__((ext_vector_type(2))) float v2f;
typedef __attribute__((ext_vector_type(4))) float v4f;
typedef __attribute__((ext_vector_type(8))) float v8f;

// ---------------- 0) zero-fill d_out (536 MB) with NT b128 stores --------
__global__ void zero_f4(v4f* __restrict__ p, long long n) {
  long long i = (long long)blockIdx.x * blockDim.x + threadIdx.x;
  const long long stride = (long long)gridDim.x * blockDim.x;
  v4f z = {0.f, 0.f, 0.f, 0.f};
  for (; i < n; i += stride) __builtin_nontemporal_store(z, p + i);
}

__global__ void zero_f1(float* __restrict__ p, long long n) {
  long long i = threadIdx.x;
  for (; i < n; i += blockDim.x) p[i] = 0.f;
}

// ---------------- 1) logits = x @ wg^T via V_WMMA_F32_16X16X4_F32 --------
// Grid: S/64 blocks of 128 threads (4 waves). Each block computes a
// 64-row x 64-expert tile. Wave w owns token rows [16w,16w+16) and all 64
// expert columns (4 accumulators). K staged in LDS in chunks of 64.
constexpr int KC = 64;

__global__ __launch_bounds__(128) void gate_gemm_wmma(
    const float* __restrict__ x, const float* __restrict__ wg,
    float* __restrict__ logits) {
  __shared__ float lds_x[64 * KC];   // [row][k]
  __shared__ float lds_w[NE * KC];   // [expert][k]

  const int tid  = threadIdx.x;      // 0..127
  const int wave = tid >> 5;         // 0..3
  const int lane = tid & 31;
  const int row0 = blockIdx.x * 64;

  v8f acc0 = {}, acc1 = {}, acc2 = {}, acc3 = {};

  const int m = lane & 15;           // tile row (A) / tile col (B)
  const int h = lane >> 4;           // K-half selector

  for (int kb = 0; kb < MD; kb += KC) {
    // cooperative staging: 64x64 x-tile + 64x64 wg-tile (16 KB each)
    for (int i = tid; i < 64 * (KC / 4); i += 128) {
      const int r  = i / (KC / 4);
      const int k4 = (i % (KC / 4)) * 4;
      const v4f v = *(const v4f*)(x + (size_t)(row0 + r) * MD + kb + k4);
      *(v4f*)(&lds_x[r * KC + k4]) = v;
    }
    for (int i = tid; i < NE * (KC / 4); i += 128) {
      const int e  = i / (KC / 4);
      const int k4 = (i % (KC / 4)) * 4;
      const v4f v = *(const v4f*)(wg + (size_t)e * MD + kb + k4);
      *(v4f*)(&lds_w[e * KC + k4]) = v;
    }
    __syncthreads();

    for (int kj = 0; kj < KC; kj += 4) {
      // A frag (16x4 f32): lane<16 holds K0,K1; lane>=16 holds K2,K3 (ISA 7.12.2)
      const v2f a = *(const v2f*)(&lds_x[(wave * 16 + m) * KC + kj + 2 * h]);
      // B frag (4x16 f32): lane = column N (expert), same K-pair split
      const v2f b0 = *(const v2f*)(&lds_w[( 0 + m) * KC + kj + 2 * h]);
      const v2f b1 = *(const v2f*)(&lds_w[(16 + m) * KC + kj + 2 * h]);
      const v2f b2 = *(const v2f*)(&lds_w[(32 + m) * KC + kj + 2 * h]);
      const v2f b3 = *(const v2f*)(&lds_w[(48 + m) * KC + kj + 2 * h]);
      acc0 = __builtin_amdgcn_wmma_f32_16x16x4_f32(false, a, false, b0, (short)0, acc0, false, false);
      acc1 = __builtin_amdgcn_wmma_f32_16x16x4_f32(false, a, false, b1, (short)0, acc1, false, false);
      acc2 = __builtin_amdgcn_wmma_f32_16x16x4_f32(false, a, false, b2, (short)0, acc2, false, false);
      acc3 = __builtin_amdgcn_wmma_f32_16x16x4_f32(false, a, false, b3, (short)0, acc3, false, false);
    }
    __syncthreads();
  }

  // C/D layout: VGPR j -> M = j (lanes 0-15) / j+8 (lanes 16-31), N = lane&15
  const int n = lane & 15;
  #pragma unroll
  for (int j = 0; j < 8; ++j) {
    const int mr = j + 8 * h;
    float* row = logits + (size_t)(row0 + wave * 16 + mr) * NE + n;
    row[ 0] = acc0[j];
    row[16] = acc1[j];
    row[32] = acc2[j];
    row[48] = acc3[j];
  }
}

// ---------------- 2) softmax + top-1 per token (one wave per token) ------
__global__ __launch_bounds__(256) void softmax_topk(
    const float* __restrict__ logits, float* __restrict__ gates,
    float* __restrict__ gate_val, int* __restrict__ expert_ix) {
  const int lane = threadIdx.x & 31;
  const int wv   = threadIdx.x >> 5;
  const int s    = blockIdx.x * 8 + wv;

  const float l0 = logits[(size_t)s * NE + lane];
  const float l1 = logits[(size_t)s * NE + lane + 32];

  float mx = fmaxf(l0, l1);
  #pragma unroll
  for (int off = 16; off >= 1; off >>= 1) mx = fmaxf(mx, __shfl_xor(mx, off, 32));

  const float p0 = expf(l0 - mx);
  const float p1 = expf(l1 - mx);
  float sum = p0 + p1;
  #pragma unroll
  for (int off = 16; off >= 1; off >>= 1) sum += __shfl_xor(sum, off, 32);
  const float inv = 1.0f / sum;

  gates[(size_t)s * NE + lane]      = p0 * inv;
  gates[(size_t)s * NE + lane + 32] = p1 * inv;

  // argmax with lowest-index tie-break (matches jnp.argmax)
  float bv = l0; int bi = lane;
  if (l1 > bv) { bv = l1; bi = lane + 32; }
  #pragma unroll
  for (int off = 16; off >= 1; off >>= 1) {
    const float ov = __shfl_xor(bv, off, 32);
    const int   oi = __shfl_xor(bi, off, 32);
    if (ov > bv || (ov == bv && oi < bi)) { bv = ov; bi = oi; }
  }
  if (lane == 0) {
    expert_ix[s] = bi;
    gate_val[s]  = expf(bv - mx) * inv;   // == gates[s, argmax]
  }
}

// ---------------- 3) per-expert rank scan + counts + me*ce partial -------
// One block per expert. Deterministic: block-scan over token order; fixed
// tree order for the gates-column reduction.
__global__ __launch_bounds__(256) void expert_rank(
    const int* __restrict__ expert_ix, const float* __restrict__ gates,
    int* __restrict__ rank, float* __restrict__ out_counts,
    float* __restrict__ partials) {
  const int e = blockIdx.x;
  const int t = threadIdx.x;
  __shared__ int   scan[256];
  __shared__ float red[256];

  int running = 0;
  float gacc = 0.f;

  for (int base = 0; base < S; base += 256) {
    const int s = base + t;
    const int flag = (expert_ix[s] == e) ? 1 : 0;
    gacc += gates[(size_t)s * NE + e];

    scan[t] = flag;
    __syncthreads();
    #pragma unroll
    for (int off = 1; off < 256; off <<= 1) {
      const int v = (t >= off) ? scan[t - off] : 0;
      __syncthreads();
      scan[t] += v;
      __syncthreads();
    }
    if (flag) rank[s] = running + scan[t] - 1;
    running += scan[255];
    __syncthreads();
  }

  red[t] = gacc;
  __syncthreads();
  #pragma unroll
  for (int off = 128; off >= 1; off >>= 1) {
    if (t < off) red[t] += red[t + off];
    __syncthreads();
  }
  if (t == 0) {
    out_counts[e] = (float)running;                              // exp_counts (pre-drop)
    const float me = red[0] / (float)S;
    const float ce = (float)running / (float)S;
    partials[e] = me * ce;
  }
}

// ---------------- 4) l_aux = E * sum(partials), one wave -----------------
__global__ void laux_kernel(const float* __restrict__ partials,
                            float* __restrict__ out_laux) {
  const int lane = threadIdx.x;  // 32 threads
  float v = partials[lane] + partials[lane + 32];
  #pragma unroll
  for (int off = 16; off >= 1; off >>= 1) v += __shfl_xor(v, off, 32);
  if (lane == 0) out_laux[0] = v * (float)NE;
}

// ---------------- 5) scatter combine_weights / dispatch_mask -------------
__global__ void scatter_out(const int* __restrict__ expert_ix,
                            const int* __restrict__ rank,
                            const float* __restrict__ gate_val,
                            float* __restrict__ out_comb,
                            float* __restrict__ out_disp) {
  const int s = blockIdx.x * blockDim.x + threadIdx.x;
  if (s >= S) return;
  const int r = rank[s];
  if (r < CAP) {
    const int e = expert_ix[s];
    const size_t off = ((size_t)s * NE + e) * CAP + r;
    out_comb[off] = gate_val[s];
    out_disp[off] = 1.0f;   // dispatch_mask = (combine != 0)
  }
}

// ---------------- launch --------------------------------------------------
extern "C" void kernel_launch(void* const* d_in, const int* in_sizes, int n_in,
                              void* d_out, int out_size, void* d_ws, size_t ws_size,
                              hipStream_t stream) {
  const float* x  = (const float*)d_in[0];
  const float* wg = (const float*)d_in[1];
  float* out = (float*)d_out;

  char* ws = (char*)d_ws;
  float* logits    = (float*)(ws);                                   // S*NE
  float* gates     = (float*)(ws + (size_t)S * NE * 4);              // S*NE
  float* gate_val  = (float*)(ws + (size_t)2 * S * NE * 4);          // S
  int*   expert_ix = (int*)  (ws + (size_t)2 * S * NE * 4 + (size_t)S * 4);
  int*   rank      = (int*)  (ws + (size_t)2 * S * NE * 4 + (size_t)2 * S * 4);
  float* partials  = (float*)(ws + (size_t)2 * S * NE * 4 + (size_t)3 * S * 4);

  // output layout: [l_aux(1)] [combine (S*NE*CAP)] [dispatch (S*NE*CAP)] [counts (NE)]
  float* out_laux = out;
  float* out_comb = out + 1;
  float* out_disp = out + 1 + SEC;
  float* out_cnts = out + 1 + 2 * SEC;

  // zero the whole output (dominant cost: 536 MB of NT b128 stores)
  const long long n4   = (long long)out_size >> 2;
  const long long tail = (long long)out_size - n4 * 4;
  zero_f4<<<2048, 256, 0, stream>>>((v4f*)out, n4);
  if (tail > 0) zero_f1<<<1, 64, 0, stream>>>(out + n4 * 4, tail);

  gate_gemm_wmma<<<S / 64, 128, 0, stream>>>(x, wg, logits);
  softmax_topk  <<<S / 8, 256, 0, stream>>>(logits, gates, gate_val, expert_ix);
  expert_rank   <<<NE, 256, 0, stream>>>(expert_ix, gates, rank, out_cnts, partials);
  laux_kernel   <<<1, 32, 0, stream>>>(partials, out_laux);
  scatter_out   <<<(S + 255) / 256, 256, 0, stream>>>(expert_ix, rank, gate_val,
                                                      out_comb, out_disp);
}